// DirectionalHMAGAT_910533067067
// MI455X (gfx1250) — compile-verified
//
#include <hip/hip_runtime.h>
#include <hip/hip_bf16.h>

typedef __attribute__((ext_vector_type(2))) float v2f;
typedef __attribute__((ext_vector_type(8))) float v8f;

// ---------------------------------------------------------------------------
// K0: out <- broadcast(bias), denom <- 0, smax_enc <- 0
// ---------------------------------------------------------------------------
__global__ void __launch_bounds__(256)
k_init(float* __restrict__ out, const float* __restrict__ bias,
       float* __restrict__ denom, unsigned int* __restrict__ smax_enc, int N_)
{
    long long idx = (long long)blockIdx.x * blockDim.x + threadIdx.x;
    long long total = (long long)N_ * 256;
    if (idx < total) out[idx] = bias[idx & 255];
    if (idx < (long long)N_ * 4) { denom[idx] = 0.0f; smax_enc[idx] = 0u; }
}

// ---------------------------------------------------------------------------
// K1: D[M,256] = A[M,64] @ B[64,256] via V_WMMA_F32_16X16X4_F32.
// One wave32 per 16x16 output tile; 16 k-steps of 16x16x4.
// f32 A operand (16x4): lanes 0-15 rows m0..m0+15 with K={k0,k0+1},
//                       lanes 16-31 same rows with K={k0+2,k0+3}.
// f32 B operand (4x16): lanes 0-15 cols n0..n0+15 rows {k0,k0+1},
//                       lanes 16-31 same cols rows {k0+2,k0+3}.
// C/D (16x16): VGPR v -> row m0 + v + (lane>=16 ? 8 : 0), col n0 + (lane&15).
// ---------------------------------------------------------------------------
__global__ void __launch_bounds__(256)
k_gemm_f32_wmma(const float* __restrict__ A,   // [M, 64]
                const float* __restrict__ B,   // [64, 256]
                float* __restrict__ D,         // [M, 256]
                int M)
{
    const int lane = threadIdx.x & 31;
    const int wave = threadIdx.x >> 5;
    const int tile = blockIdx.x * 8 + wave;
    const int mt = tile >> 4;          // 16 column tiles (256/16)
    const int nt = tile & 15;
    const int m0 = mt * 16;
    const int n0 = nt * 16;
    if (m0 >= M) return;               // wave-uniform: EXEC stays all-ones

    const int half = lane >> 4;        // 0: K lo pair, 1: K hi pair
    const int l16  = lane & 15;
    int mrow = m0 + l16; if (mrow >= M) mrow = M - 1;   // clamp (reads only)
    const float* Arow = A + (long long)mrow * 64;

    v8f acc = {};
#pragma unroll
    for (int k0 = 0; k0 < 64; k0 += 4) {
        const int ka = k0 + 2 * half;
        float2 av = *(const float2*)(Arow + ka);
        v2f a; a.x = av.x; a.y = av.y;
        v2f b;
        b.x = B[(ka + 0) * 256 + n0 + l16];
        b.y = B[(ka + 1) * 256 + n0 + l16];
        acc = __builtin_amdgcn_wmma_f32_16x16x4_f32(
            /*neg_a=*/false, a, /*neg_b=*/false, b,
            /*c_mod=*/(short)0, acc, /*reuse_a=*/false, /*reuse_b=*/false);
    }

    const int rbase = m0 + half * 8;
#pragma unroll
    for (int v = 0; v < 8; ++v) {
        int r = rbase + v;
        if (r < M) D[(long long)r * 256 + n0 + l16] = acc[v];
    }
}

// ---------------------------------------------------------------------------
// K2: per-edge attention scores. Half-wave (16 lanes) per edge.
// score[e,h] = leaky_relu( dot(u[src,h,:], x[dst,:]) / 8 , 0.2 )
// Also atomic-max into smax_enc[dst,h] (order-preserving uint encoding).
// ---------------------------------------------------------------------------
__device__ __forceinline__ unsigned int enc_f32(float f) {
    unsigned int b = __float_as_uint(f);
    return (b & 0x80000000u) ? ~b : (b | 0x80000000u);
}
__device__ __forceinline__ float dec_f32(unsigned int e) {
    return __uint_as_float((e & 0x80000000u) ? (e ^ 0x80000000u) : ~e);
}

__global__ void __launch_bounds__(256)
k_edge_score(const float* __restrict__ x,      // [N,64]
             const float* __restrict__ u,      // [N,256]
             const int* __restrict__ ei,       // [2,E]
             float* __restrict__ score,        // [E,4]
             unsigned int* __restrict__ smax_enc, // [N,4]
             int E_)
{
    const int sub = threadIdx.x >> 4;
    const int l   = threadIdx.x & 15;
    const int e   = blockIdx.x * 16 + sub;
    if (e >= E_) return;
    const int src = ei[e];
    const int dst = ei[E_ + e];

    const float4 xd = *(const float4*)(x + (long long)dst * 64 + l * 4);
#pragma unroll
    for (int h = 0; h < 4; ++h) {
        const float4 uv = *(const float4*)(u + (long long)src * 256 + h * 64 + l * 4);
        float p = uv.x * xd.x + uv.y * xd.y + uv.z * xd.z + uv.w * xd.w;
        p += __shfl_xor(p, 8, 16);
        p += __shfl_xor(p, 4, 16);
        p += __shfl_xor(p, 2, 16);
        p += __shfl_xor(p, 1, 16);
        if (l == 0) {
            float s = p * 0.125f;                 // / sqrt(64)
            s = (s > 0.0f) ? s : 0.2f * s;        // leaky relu
            score[e * 4 + h] = s;
            atomicMax(&smax_enc[dst * 4 + h], enc_f32(s));
        }
    }
}

// ---------------------------------------------------------------------------
// K3: e = exp(score - smax[dst]) * edge_weight; accumulate denom[dst,h].
// score buffer is overwritten in-place with e.
// ---------------------------------------------------------------------------
__global__ void __launch_bounds__(256)
k_edge_exp(const int* __restrict__ ei, const float* __restrict__ ew,
           const unsigned int* __restrict__ smax_enc,
           float* __restrict__ score, float* __restrict__ denom, int E_)
{
    int idx = blockIdx.x * blockDim.x + threadIdx.x;
    if (idx >= E_ * 4) return;
    int e = idx >> 2, h = idx & 3;
    int dst = ei[E_ + e];
    float smax = dec_f32(smax_enc[dst * 4 + h]);
    float v = expf(score[idx] - smax) * ew[e];
    score[idx] = v;
    atomicAdd(&denom[dst * 4 + h], v);
}

// ---------------------------------------------------------------------------
// K4: wave per edge: alpha = e / (denom[dst]+1e-16);
//     out[dst, :] += alpha[h] * xs[src, :]   (256 channels, 8 per lane)
// ---------------------------------------------------------------------------
__global__ void __launch_bounds__(256)
k_edge_aggregate(const int* __restrict__ ei,
                 const float* __restrict__ eBuf,   // [E,4]
                 const float* __restrict__ denom,  // [N,4]
                 const float* __restrict__ xs,     // [N,256]
                 float* __restrict__ out,          // [N,256]
                 int E_)
{
    const int lane = threadIdx.x & 31;
    const int wid  = threadIdx.x >> 5;
    const int e    = blockIdx.x * 8 + wid;
    if (e >= E_) return;
    const int src = ei[e];
    const int dst = ei[E_ + e];

    float alpha[4];
#pragma unroll
    for (int h = 0; h < 4; ++h)
        alpha[h] = eBuf[e * 4 + h] / (denom[dst * 4 + h] + 1e-16f);

    const float* xrow = xs + (long long)src * 256;
    float* orow = out + (long long)dst * 256;
#pragma unroll
    for (int i = 0; i < 8; ++i) {
        int c = i * 32 + lane;
        atomicAdd(&orow[c], alpha[c >> 6] * xrow[c]);
    }
}

// ---------------------------------------------------------------------------
extern "C" void kernel_launch(void* const* d_in, const int* in_sizes, int n_in,
                              void* d_out, int out_size, void* d_ws, size_t ws_size,
                              hipStream_t stream) {
    const float* x    = (const float*)d_in[0];
    const int*   ei   = (const int*)d_in[1];
    const float* ew   = (const float*)d_in[2];
    const float* Wlin = (const float*)d_in[3];
    const float* Watt = (const float*)d_in[4];
    const float* bias = (const float*)d_in[5];
    float* out = (float*)d_out;

    const int N_ = in_sizes[0] / 64;   // 50000
    const int E_ = in_sizes[1] / 2;    // 800000

    // workspace carve-out (256B aligned)
    char* ws = (char*)d_ws;
    size_t off = 0;
    auto carve = [&](size_t bytes) { size_t p = off; off += (bytes + 255) & ~(size_t)255; return p; };
    float*        xs    = (float*)(ws + carve((size_t)N_ * 256 * 4));
    float*        u     = (float*)(ws + carve((size_t)N_ * 256 * 4));
    float*        sc    = (float*)(ws + carve((size_t)E_ * 4 * 4));
    unsigned int* smax  = (unsigned int*)(ws + carve((size_t)N_ * 4 * 4));
    float*        denom = (float*)(ws + carve((size_t)N_ * 4 * 4));

    // K0: init out=bias, denom=0, smax_enc=0
    {
        long long total = (long long)N_ * 256;
        int blocks = (int)((total + 255) / 256);
        k_init<<<blocks, 256, 0, stream>>>(out, bias, denom, smax, N_);
    }
    // K1: xs = x @ W_lin ; u = x @ W_att  (WMMA f32 16x16x4)
    {
        int mtiles = (N_ + 15) / 16;
        int tiles  = mtiles * 16;            // 16 column tiles
        int blocks = (tiles + 7) / 8;        // 8 waves per block
        k_gemm_f32_wmma<<<blocks, 256, 0, stream>>>(x, Wlin, xs, N_);
        k_gemm_f32_wmma<<<blocks, 256, 0, stream>>>(x, Watt, u,  N_);
    }
    // K2: scores + segment max
    {
        int blocks = (E_ + 15) / 16;
        k_edge_score<<<blocks, 256, 0, stream>>>(x, u, ei, sc, smax, E_);
    }
    // K3: exp + denom
    {
        int blocks = (E_ * 4 + 255) / 256;
        k_edge_exp<<<blocks, 256, 0, stream>>>(ei, ew, smax, sc, denom, E_);
    }
    // K4: weighted scatter-add of messages
    {
        int blocks = (E_ + 7) / 8;
        k_edge_aggregate<<<blocks, 256, 0, stream>>>(ei, sc, denom, xs, out, E_);
    }
}